// LatentLinearModel_19344532702169
// MI455X (gfx1250) — compile-verified
//
#include <hip/hip_runtime.h>

// CDNA5 / gfx1250. wave32.
typedef __attribute__((ext_vector_type(2))) float v2f;
typedef __attribute__((ext_vector_type(8))) float v8f;

#define KDIM 64

// One wave32 processes a tile of 16 rows.
//   Gather layout (matches the native V_WMMA_F32_16X16X4_F32 A layout):
//     lane L (L<16):  row = L,      K sub-chunk {0,1} of each 4-wide K step
//     lane L (L>=16): row = L-16,   K sub-chunk {2,3}
//   Per K step: A = elementwise u*v products (2 VGPRs), B = all-ones 4x16,
//   so D[m][n] += sum_k u[m,k]*v[m,k]  (every column n holds the row dot).
__global__ __launch_bounds__(256) void latent_dot_wmma_kernel(
    const int* __restrict__ users, const int* __restrict__ jokes,
    const float* __restrict__ U, const float* __restrict__ V,
    const float* __restrict__ a, const float* __restrict__ b,
    const float* __restrict__ g, float* __restrict__ out, int Btot)
{
    const int lane = threadIdx.x & 31;
    const int wave = threadIdx.x >> 5;
    const int tile = blockIdx.x * 8 + wave;
    const int base = tile * 16;
    if (base >= Btot) return;               // wave-uniform exit, EXEC stays all-1s

    const int rowWant = base + (lane & 15);
    const int row     = (rowWant < Btot) ? rowWant : (Btot - 1);   // clamp, keep EXEC full
    const int usr     = users[row];
    const int jok     = jokes[row];

    // K sub-offset inside each 4-wide K step: lanes 0-15 -> +0, lanes 16-31 -> +2
    const int ksub = (lane >> 4) << 1;
    const float* __restrict__ up = U + (size_t)usr * KDIM + ksub;
    const float* __restrict__ vp = V + (size_t)jok * KDIM + ksub;

    // Bias gather early to overlap latency. U/a are streamed once -> non-temporal
    // (keeps the 25.6MB V table resident in the 192MB L2). b/V/g are reused -> RT.
    const float bias = __builtin_nontemporal_load(a + usr) + b[jok] + g[0];

    v2f ones; ones.x = 1.0f; ones.y = 1.0f;   // B matrix = all ones (any layout)
    v8f c = {};                                // 16x16 f32 accumulator (8 VGPRs)

#pragma unroll
    for (int k = 0; k < KDIM; k += 4) {
        v2f uu = __builtin_nontemporal_load(reinterpret_cast<const v2f*>(up + k));
        v2f vv = *reinterpret_cast<const v2f*>(vp + k);
        v2f prod; prod.x = uu.x * vv.x; prod.y = uu.y * vv.y;
        // D = A(16x4 products) * B(4x16 ones) + C
        c = __builtin_amdgcn_wmma_f32_16x16x4_f32(
                /*neg_a=*/false, prod, /*neg_b=*/false, ones,
                /*c_mod=*/(short)0, c, /*reuse_a=*/false, /*reuse_b=*/false);
    }

    // Extract: lane L in [0,8) holds rows 0..7 at VGPR r (M=r); lanes [24,32)
    // hold rows 8..15 at VGPR r (M=8+r). For both groups M == (lane & 15) when
    // r == (lane & 7), i.e. each active lane's own gathered row index matches
    // the output row it emits. All 16 columns are identical, so N is irrelevant.
    const int idx = lane & 7;
    float s01 = (idx & 1) ? c[1] : c[0];
    float s23 = (idx & 1) ? c[3] : c[2];
    float s45 = (idx & 1) ? c[5] : c[4];
    float s67 = (idx & 1) ? c[7] : c[6];
    float s03 = (idx & 2) ? s23 : s01;
    float s47 = (idx & 2) ? s67 : s45;
    float dot = (idx & 4) ? s47 : s03;

    const bool active = (lane < 8) || (lane >= 24);
    if (active && rowWant < Btot) {
        out[row] = dot + bias;   // lanes 0-7 -> rows base+0..7, lanes 24-31 -> base+8..15
    }
}

extern "C" void kernel_launch(void* const* d_in, const int* in_sizes, int n_in,
                              void* d_out, int out_size, void* d_ws, size_t ws_size,
                              hipStream_t stream) {
    const int*   users = (const int*)  d_in[0];
    const int*   jokes = (const int*)  d_in[1];
    const float* U     = (const float*)d_in[2];
    const float* V     = (const float*)d_in[3];
    const float* a     = (const float*)d_in[4];
    const float* b     = (const float*)d_in[5];
    const float* g     = (const float*)d_in[6];
    float*       out   = (float*)d_out;
    const int Btot = in_sizes[0];

    const int tiles  = (Btot + 15) / 16;   // 16 rows per wave32
    const int blocks = (tiles + 7) / 8;    // 8 waves per 256-thread block
    latent_dot_wmma_kernel<<<dim3(blocks), dim3(256), 0, stream>>>(
        users, jokes, U, V, a, b, g, out, Btot);
}